// HGT_10943576670218
// MI455X (gfx1250) — compile-verified
//
#include <hip/hip_runtime.h>
#include <hip/hip_bf16.h>
#include <math.h>

#define H 128

typedef __attribute__((ext_vector_type(2))) float v2f;
typedef __attribute__((ext_vector_type(8))) float v8f;

// ---------------------------------------------------------------- helpers

__device__ __forceinline__ float gelu_exact(float x) {
    return 0.5f * x * (1.0f + erff(x * 0.70710678118654752440f));
}

// monotone float<->uint mapping so float-max can use hardware uint atomicMax
__device__ __forceinline__ unsigned f2ord(float f) {
    unsigned u = __float_as_uint(f);
    return (u & 0x80000000u) ? ~u : (u | 0x80000000u);
}
__device__ __forceinline__ float ord2f(unsigned u) {
    unsigned v = (u & 0x80000000u) ? (u ^ 0x80000000u) : ~u;
    return __uint_as_float(v);
}
#define ORD_NEG_INF 0x007FFFFFu   // f2ord(-inf)

__device__ __forceinline__ void atomAddF(float* p, float v) {
    unsafeAtomicAdd(p, v);   // lowers to global_atomic_add_f32
}

// ---------------------------------------------------------------- WMMA GEMM
// C[N x 128] = op(A[N x K]) @ W[K x 128] + bias, with optional skip-mix+relu.
// 256 threads = 8 waves; block handles a 128-row strip; each wave produces a
// 16x128 strip as 8 WMMA 16x16 tiles using v_wmma_f32_16x16x4_f32 (fp32 A/B).
template<int K, bool PRE_GELU, bool MIX_RELU>
__global__ __launch_bounds__(256) void gemm_wmma_k(
    const float* __restrict__ A,      // N x K row-major
    const float* __restrict__ W,      // K x 128 row-major
    const float* __restrict__ bias,   // 128 (may be null)
    float* __restrict__ C,            // N x 128 row-major
    const float* __restrict__ xold,   // MIX_RELU: skip-connection input
    const float* __restrict__ skip,   // MIX_RELU: scalar; beta = sigmoid(*skip)
    int N)
{
    constexpr int LDK = K + 2;        // pad: conflict-free ds_load_b64
    extern __shared__ float smem[];
    float* lA = smem;                  // 128 * LDK   (row-major A strip)
    float* lW = smem + 128 * LDK;      // 128 * LDK   (W transposed: [col][k])

    const int tid = threadIdx.x;
    const int rowBase = blockIdx.x * 128;

    // stage A strip (optionally pre-GELU), zero-fill rows past N
    for (int idx = tid; idx < 128 * K / 4; idx += 256) {
        int r = (idx * 4) / K, k = (idx * 4) % K;
        float4 v4 = make_float4(0.f, 0.f, 0.f, 0.f);
        int gr = rowBase + r;
        if (gr < N) v4 = *(const float4*)(A + (size_t)gr * K + k);
        if (PRE_GELU) {
            v4.x = gelu_exact(v4.x); v4.y = gelu_exact(v4.y);
            v4.z = gelu_exact(v4.z); v4.w = gelu_exact(v4.w);
        }
        float* p = lA + r * LDK + k;
        p[0] = v4.x; p[1] = v4.y; p[2] = v4.z; p[3] = v4.w;
    }
    // stage W transposed so B fragments are contiguous pairs along K
    for (int idx = tid; idx < K * 128; idx += 256) {
        int k = idx >> 7, c = idx & 127;
        lW[c * LDK + k] = W[idx];
    }
    __syncthreads();

    const int wid  = tid >> 5;
    const int lane = tid & 31;
    const int l16  = lane & 15;
    const int koff = (lane >> 4) << 1;     // lanes 16-31 hold K pair +2
    const int rowL = wid * 16 + l16;

    v8f acc[8];
    #pragma unroll
    for (int t = 0; t < 8; ++t)
        acc[t] = (v8f){0.f, 0.f, 0.f, 0.f, 0.f, 0.f, 0.f, 0.f};

    for (int k0 = 0; k0 < K; k0 += 4) {
        v2f a = *(const v2f*)(lA + rowL * LDK + k0 + koff);
        #pragma unroll
        for (int tn = 0; tn < 8; ++tn) {
            v2f b = *(const v2f*)(lW + (tn * 16 + l16) * LDK + k0 + koff);
            acc[tn] = __builtin_amdgcn_wmma_f32_16x16x4_f32(
                false, a, false, b, (short)0, acc[tn], false, false);
        }
    }

    float beta = 1.0f, ombeta = 0.0f;
    if (MIX_RELU) {
        float s = skip[0];
        beta = 1.0f / (1.0f + expf(-s));
        ombeta = 1.0f - beta;
    }

    // C/D layout: vgpr i -> row (i + 8*(lane>=16)), col = lane&15
    #pragma unroll
    for (int tn = 0; tn < 8; ++tn) {
        int col = tn * 16 + l16;
        float bb = bias ? bias[col] : 0.0f;
        #pragma unroll
        for (int i = 0; i < 8; ++i) {
            int r = rowBase + wid * 16 + i + ((lane >> 4) << 3);
            if (r < N) {
                float v = acc[tn][i] + bb;
                if (MIX_RELU) {
                    v = beta * v + ombeta * xold[(size_t)r * H + col];
                    v = fmaxf(v, 0.0f);
                }
                C[(size_t)r * H + col] = v;
            }
        }
    }
}

// ---------------------------------------------------------------- edge phase

// wave per edge: 128-dim dot(q[dst], ka[src]) -> score; atomic segment max
__global__ __launch_bounds__(256) void edge_score(
    const float* __restrict__ q, const float* __restrict__ ka,
    const int* __restrict__ src, const int* __restrict__ dst,
    const float* __restrict__ p_rel, float* __restrict__ sc,
    unsigned* __restrict__ mbuf, int E)
{
    int e = blockIdx.x * 8 + (threadIdx.x >> 5);
    int lane = threadIdx.x & 31;
    if (e >= E) return;
    int s = src[e], d = dst[e];
    float4 qa = *(const float4*)(q  + (size_t)d * H + lane * 4);
    float4 kb = *(const float4*)(ka + (size_t)s * H + lane * 4);
    float v = qa.x * kb.x + qa.y * kb.y + qa.z * kb.z + qa.w * kb.w;
    #pragma unroll
    for (int off = 16; off; off >>= 1) v += __shfl_down(v, off, 32);
    if (lane == 0) {
        float a = v * p_rel[0] * 0.08838834764831845f;  // 1/sqrt(128)
        sc[e] = a;
        atomicMax(mbuf + d, f2ord(a));
    }
}

// thread per edge: e = exp(a - m[dst]); z[dst] += e
__global__ __launch_bounds__(256) void edge_exp(
    float* __restrict__ sc, const int* __restrict__ dst,
    const unsigned* __restrict__ mbuf, float* __restrict__ z, int E)
{
    int e = blockIdx.x * 256 + threadIdx.x;
    if (e >= E) return;
    int d = dst[e];
    float ev = expf(sc[e] - ord2f(mbuf[d]));
    sc[e] = ev;
    atomAddF(z + d, ev);
}

// wave per edge: agg[dst] += (e/z[dst]) * vm[src]  (128-wide scatter-add)
__global__ __launch_bounds__(256) void edge_scatter(
    const float* __restrict__ sc, const float* __restrict__ z,
    const float* __restrict__ vm, const int* __restrict__ src,
    const int* __restrict__ dst, float* __restrict__ agg, int E)
{
    int e = blockIdx.x * 8 + (threadIdx.x >> 5);
    int lane = threadIdx.x & 31;
    if (e >= E) return;
    int s = src[e], d = dst[e];
    float coef = sc[e] / (z[d] + 1e-16f);
    float4 v4 = *(const float4*)(vm + (size_t)s * H + lane * 4);
    float* ag = agg + (size_t)d * H + lane * 4;
    atomAddF(ag + 0, coef * v4.x);
    atomAddF(ag + 1, coef * v4.y);
    atomAddF(ag + 2, coef * v4.z);
    atomAddF(ag + 3, coef * v4.w);
}

// ---------------------------------------------------------------- small kernels

__global__ void vecmat128(const float* __restrict__ vec,
                          const float* __restrict__ W, float* __restrict__ out) {
    int n = threadIdx.x;   // 128 threads
    float acc = 0.f;
    for (int k = 0; k < H; ++k) acc += vec[k] * W[k * H + n];
    out[n] = acc;
}

__global__ void fill_f32(float* p, float v, int n) {
    int i = blockIdx.x * 256 + threadIdx.x;
    if (i < n) p[i] = v;
}
__global__ void fill_u32(unsigned* p, unsigned v, int n) {
    int i = blockIdx.x * 256 + threadIdx.x;
    if (i < n) p[i] = v;
}

__global__ void gather128(const float* __restrict__ emb, const int* __restrict__ ids,
                          float* __restrict__ out, int n) {
    int i = blockIdx.x * 256 + threadIdx.x;   // over n*32 float4s
    if (i < n * 32) {
        int row = i >> 5, c = i & 31;
        ((float4*)out)[i] = ((const float4*)(emb + (size_t)ids[row] * H))[c];
    }
}

__global__ __launch_bounds__(256) void out_linear(
    const float* __restrict__ x, const float* __restrict__ w,
    const float* __restrict__ b, float* __restrict__ out, int N)
{
    __shared__ float lw[256];                 // 128 x 2
    lw[threadIdx.x] = w[threadIdx.x];
    __syncthreads();
    int r = blockIdx.x * 256 + threadIdx.x;
    if (r >= N) return;
    float a0 = b[0], a1 = b[1];
    const float4* xr = (const float4*)(x + (size_t)r * H);
    #pragma unroll 8
    for (int c = 0; c < 32; ++c) {
        float4 v = xr[c];
        int k = c * 4;
        a0 += v.x * lw[(k + 0) * 2] + v.y * lw[(k + 1) * 2]
            + v.z * lw[(k + 2) * 2] + v.w * lw[(k + 3) * 2];
        a1 += v.x * lw[(k + 0) * 2 + 1] + v.y * lw[(k + 1) * 2 + 1]
            + v.z * lw[(k + 2) * 2 + 1] + v.w * lw[(k + 3) * 2 + 1];
    }
    out[r * 2 + 0] = a0;
    out[r * 2 + 1] = a1;
}

// ---------------------------------------------------------------- driver

extern "C" void kernel_launch(void* const* d_in, const int* in_sizes, int n_in,
                              void* d_out, int out_size, void* d_ws, size_t ws_size,
                              hipStream_t stream)
{
    (void)n_in; (void)out_size; (void)ws_size;

    const float* x_tx       = (const float*)d_in[0];
    const float* x_cust     = (const float*)d_in[1];
    const int*   merch_ids  = (const int*)  d_in[2];
    const int*   e_ct_src   = (const int*)  d_in[3];
    const int*   e_ct_dst   = (const int*)  d_in[4];
    const int*   e_mt_src   = (const int*)  d_in[5];
    const int*   e_mt_dst   = (const int*)  d_in[6];
    const float* proj_tx_w  = (const float*)d_in[7];
    const float* proj_tx_b  = (const float*)d_in[8];
    const float* proj_cu_w  = (const float*)d_in[9];
    const float* proj_cu_b  = (const float*)d_in[10];
    const float* merch_emb  = (const float*)d_in[11];
    const float* kw  = (const float*)d_in[12];
    const float* kb  = (const float*)d_in[13];
    const float* qw  = (const float*)d_in[14];
    const float* qb  = (const float*)d_in[15];
    const float* vw  = (const float*)d_in[16];
    const float* vb  = (const float*)d_in[17];
    const float* aw  = (const float*)d_in[18];
    const float* ab  = (const float*)d_in[19];
    const float* skip  = (const float*)d_in[20];
    const float* a_rel = (const float*)d_in[21];
    const float* m_rel = (const float*)d_in[22];
    const float* p_rel = (const float*)d_in[23];
    const float* out_w = (const float*)d_in[24];
    const float* out_b = (const float*)d_in[25];
    float* logits = (float*)d_out;

    const int NT = in_sizes[0] / 16;
    const int NC = in_sizes[1] / 32;
    const int NM = in_sizes[2];
    const int NE = in_sizes[3];
    const int Ls = 2;
    const int sizes[3] = {NT, NC, NM};
    const int EST[4] = {1, 2, 0, 0};
    const int EDT[4] = {0, 0, 1, 2};
    const int* esrc[4] = {e_ct_src, e_mt_src, e_ct_dst, e_mt_dst};
    const int* edst[4] = {e_ct_dst, e_mt_dst, e_ct_src, e_mt_src};

    // ---- workspace partition
    float* ws = (float*)d_ws;
    size_t off = 0;
    auto alloc = [&](size_t n) { float* p = ws + off; off += n; return p; };
    float* X[3];  for (int t = 0; t < 3; ++t) X[t]  = alloc((size_t)sizes[t] * H);
    float* Q[3];  for (int t = 0; t < 3; ++t) Q[t]  = alloc((size_t)sizes[t] * H);
    float* AG[3]; for (int t = 0; t < 3; ++t) AG[t] = alloc((size_t)sizes[t] * H);
    float*    KA  = alloc((size_t)NT * H);
    float*    VM  = alloc((size_t)NT * H);
    float*    SC  = alloc((size_t)NE);
    unsigned* Mb  = (unsigned*)alloc((size_t)NT);
    float*    Z   = alloc((size_t)NT);
    float*    WKA = alloc((size_t)Ls * 4 * H * H);
    float*    WVM = alloc((size_t)Ls * 4 * H * H);
    float*    BKA = alloc((size_t)Ls * 4 * H);
    float*    BVM = alloc((size_t)Ls * 4 * H);
    const size_t totAgg = (size_t)(NT + NC + NM) * H;   // AG[0..2] contiguous

    const int SM128 = 2 * 128 * (128 + 2) * 4;
    const int SM32  = 2 * 128 * (32 + 2) * 4;
    const int SM16  = 2 * 128 * (16 + 2) * 4;

    auto gemm = [&](const float* A, const float* Wm, const float* bias,
                    float* C, int N) {
        dim3 g((N + 127) / 128);
        gemm_wmma_k<128, false, false><<<g, 256, SM128, stream>>>(
            A, Wm, bias, C, nullptr, nullptr, N);
    };

    // ---- 1. fused relation weights: WKA = kw[st]@a_rel[r], WVM = vw[st]@m_rel[r]
    for (int l = 0; l < Ls; ++l)
        for (int r = 0; r < 4; ++r) {
            int st = EST[r];
            const float* Ar = a_rel + ((size_t)l * 4 + r) * H * H;
            const float* Mr = m_rel + ((size_t)l * 4 + r) * H * H;
            float* wka = WKA + ((size_t)l * 4 + r) * H * H;
            float* wvm = WVM + ((size_t)l * 4 + r) * H * H;
            gemm(kw + ((size_t)l * 3 + st) * H * H, Ar, nullptr, wka, H);
            gemm(vw + ((size_t)l * 3 + st) * H * H, Mr, nullptr, wvm, H);
            vecmat128<<<1, 128, 0, stream>>>(kb + ((size_t)l * 3 + st) * H, Ar,
                                             BKA + ((size_t)l * 4 + r) * H);
            vecmat128<<<1, 128, 0, stream>>>(vb + ((size_t)l * 3 + st) * H, Mr,
                                             BVM + ((size_t)l * 4 + r) * H);
        }

    // ---- 2. input projections / embedding gather
    gemm_wmma_k<16, false, false><<<dim3((NT + 127) / 128), 256, SM16, stream>>>(
        x_tx, proj_tx_w, proj_tx_b, X[0], nullptr, nullptr, NT);
    gemm_wmma_k<32, false, false><<<dim3((NC + 127) / 128), 256, SM32, stream>>>(
        x_cust, proj_cu_w, proj_cu_b, X[1], nullptr, nullptr, NC);
    gather128<<<dim3((NM * 32 + 255) / 256), 256, 0, stream>>>(
        merch_emb, merch_ids, X[2], NM);

    // ---- 3. HGT layers
    for (int l = 0; l < Ls; ++l) {
        for (int t = 0; t < 3; ++t)
            gemm(X[t], qw + ((size_t)l * 3 + t) * H * H,
                 qb + ((size_t)l * 3 + t) * H, Q[t], sizes[t]);

        fill_f32<<<dim3((int)((totAgg + 255) / 256)), 256, 0, stream>>>(
            AG[0], 0.0f, (int)totAgg);

        for (int r = 0; r < 4; ++r) {
            int st = EST[r], dt = EDT[r];
            int Ns = sizes[st], Nd = sizes[dt];
            gemm(X[st], WKA + ((size_t)l * 4 + r) * H * H,
                 BKA + ((size_t)l * 4 + r) * H, KA, Ns);
            gemm(X[st], WVM + ((size_t)l * 4 + r) * H * H,
                 BVM + ((size_t)l * 4 + r) * H, VM, Ns);
            fill_u32<<<dim3((Nd + 255) / 256), 256, 0, stream>>>(Mb, ORD_NEG_INF, Nd);
            fill_f32<<<dim3((Nd + 255) / 256), 256, 0, stream>>>(Z, 0.0f, Nd);
            edge_score<<<dim3((NE + 7) / 8), 256, 0, stream>>>(
                Q[dt], KA, esrc[r], edst[r], p_rel + (size_t)l * 4 + r, SC, Mb, NE);
            edge_exp<<<dim3((NE + 255) / 256), 256, 0, stream>>>(
                SC, edst[r], Mb, Z, NE);
            edge_scatter<<<dim3((NE + 7) / 8), 256, 0, stream>>>(
                SC, Z, VM, esrc[r], edst[r], AG[dt], NE);
        }

        // out projection: x = relu( beta*(gelu(agg)@aw + ab) + (1-beta)*x )
        for (int t = 0; t < 3; ++t)
            gemm_wmma_k<128, true, true>
                <<<dim3((sizes[t] + 127) / 128), 256, SM128, stream>>>(
                    AG[t], aw + ((size_t)l * 3 + t) * H * H,
                    ab + ((size_t)l * 3 + t) * H, X[t], X[t],
                    skip + (size_t)l * 3 + t, sizes[t]);
    }

    // ---- 4. classifier head on transaction nodes
    out_linear<<<dim3((NT + 255) / 256), 256, 0, stream>>>(
        X[0], out_w, out_b, logits, NT);
}